// SingleQMSANHead_23295902613640
// MI455X (gfx1250) — compile-verified
//
#include <hip/hip_runtime.h>
#include <math.h>

// =====================================================================
// SingleQMSANHead for MI455X (gfx1250), wave32 + WMMA f32.
//
//   K1 qkv_kernel : x @ {Wq,Wk,Wv}^T -> Q,K,V [4096,10]          (tiny)
//   K2 rho_kernel : per position build Psi (32x32 cplx) in LDS,
//                   rho = Psi Psi^H, emit 2048-long real feature
//                   vector (Re||Im) in K-packed layout FT4[k/4][m][4]
//   K3 attn_kernel: fidelity GEMM  Fq[1024x2048] . Fk^T  per batch
//                   via V_WMMA_F32_16X16X4_F32. Each wave owns a
//                   16x64 strip (4 accumulators share one A fragment
//                   -> 1.25 loads/WMMA). Manual 2-stage ping-pong
//                   K loop (#pragma unroll 1) overlaps loads of step
//                   k+1 with the 4 WMMAs of step k. __launch_bounds__
//                   (256,1) relaxes the occupancy-driven VGPR budget
//                   so the 80-register attn@V accumulators stay
//                   resident (no scratch spills). Fused causal mask,
//                   L1 row-normalize, attn@V epilogue; fixed-order
//                   deterministic reductions, no atomics.
//
// 64 MB of features lives in d_ws and is L2-resident (192 MB L2);
// the 17.2 GFLOP fp32 fidelity GEMM is the compute-bound core.
// =====================================================================

typedef __attribute__((ext_vector_type(2))) float v2f;
typedef __attribute__((ext_vector_type(8))) float v8f;

#define B_  4
#define S_  1024
#define E_  768
#define H_  10
#define DA  32
#define KF  2048            // feature vector length (Re 1024 + Im 1024)
#define KB  (KF / 4)        // 512 k-blocks of 4 for WMMA 16x16x4
#define BATCH_F (KF * S_)   // floats per batch in a feature array
#define CH  4               // j-tiles (16 cols each) per wave chunk

// workspace layout (in floats)
#define OFF_Q  0
#define OFF_K  (B_ * S_ * H_)          // 40960
#define OFF_V  (2 * B_ * S_ * H_)      // 81920
#define OFF_FQ 131072
#define OFF_FK (OFF_FQ + B_ * BATCH_F) // 131072 + 8388608
// total ws use: 16,908,288 floats = ~67.6 MB

#define WMMA_F32(A, Bv, C) \
  __builtin_amdgcn_wmma_f32_16x16x4_f32(false, (A), false, (Bv), (short)0, (C), false, false)

// ---------------------------------------------------------------------
// K1: QKV projection. One wave per position; lanes stride over E.
// ---------------------------------------------------------------------
__global__ void __launch_bounds__(128, 1)
qkv_kernel(const float* __restrict__ x,
           const float* __restrict__ Wq,
           const float* __restrict__ Wk,
           const float* __restrict__ Wv,
           float* __restrict__ ws) {
  const int wave = threadIdx.x >> 5;
  const int lane = threadIdx.x & 31;
  const int p = blockIdx.x * (blockDim.x >> 5) + wave;   // 0..4095
  const float* xr = x + (size_t)p * E_;

  float acc[3][H_];
#pragma unroll
  for (int j = 0; j < 3; j++)
#pragma unroll
    for (int h = 0; h < H_; h++) acc[j][h] = 0.0f;

  for (int e = lane; e < E_; e += 32) {
    float xv = xr[e];
#pragma unroll
    for (int h = 0; h < H_; h++) {
      acc[0][h] = fmaf(xv, Wq[h * E_ + e], acc[0][h]);
      acc[1][h] = fmaf(xv, Wk[h * E_ + e], acc[1][h]);
      acc[2][h] = fmaf(xv, Wv[h * E_ + e], acc[2][h]);
    }
  }
  // wave32 butterfly reduce
#pragma unroll
  for (int j = 0; j < 3; j++)
#pragma unroll
    for (int h = 0; h < H_; h++)
#pragma unroll
      for (int m = 16; m >= 1; m >>= 1)
        acc[j][h] += __shfl_xor(acc[j][h], m, 32);

  if (lane == 0) {
    float* Q = ws + OFF_Q;
    float* K = ws + OFF_K;
    float* V = ws + OFF_V;
#pragma unroll
    for (int h = 0; h < H_; h++) {
      Q[p * H_ + h] = acc[0][h];
      K[p * H_ + h] = acc[1][h];
      V[p * H_ + h] = acc[2][h];
    }
  }
}

// ---------------------------------------------------------------------
// K2: density-matrix feature builder. One wave per position.
// lane = column c of Psi; computes 32 sincos, stages Psi in padded LDS,
// then lane l produces rho row a=l (32 complex dots of length 32) and
// scatters it into the K-packed feature array FT4[k>>2][m][k&3].
// ---------------------------------------------------------------------
#define WPB_B 4
__global__ void __launch_bounds__(32 * WPB_B, 1)
rho_kernel(const float* __restrict__ QK,
           float* __restrict__ FT) {
  __shared__ float2 psi[WPB_B][DA * 33];   // stride 33: conflict-free
  const int wave = threadIdx.x >> 5;
  const int lane = threadIdx.x & 31;       // column c; later row a
  const int p = blockIdx.x * WPB_B + wave; // global position
  const int b = p >> 10;
  const int m = p & (S_ - 1);

  float h[H_];
#pragma unroll
  for (int i = 0; i < H_; i++) h[i] = QK[p * H_ + i];

  // T_B from low-5 wires (wire i -> bit 9-i of z, i = 5..9)
  float tb = 0.0f;
#pragma unroll
  for (int i = 5; i < H_; i++)
    tb += h[i] * (1.0f - 2.0f * (float)((lane >> (9 - i)) & 1));

#pragma unroll
  for (int a = 0; a < DA; a++) {
    float ta = 0.0f;
#pragma unroll
    for (int i = 0; i < 5; i++)
      ta += h[i] * (1.0f - 2.0f * (float)((a >> (4 - i)) & 1));
    float t   = ta + tb;
    float phi = -(0.5f * t + 0.25f * t * t);
    float sn, cs;
    __sincosf(phi, &sn, &cs);
    psi[wave][a * 33 + lane] = make_float2(cs, sn);  // e^{i phi}
  }
  __syncthreads();

  // own row a = lane
  float qr[DA], qi[DA];
#pragma unroll
  for (int c = 0; c < DA; c++) {
    float2 v = psi[wave][lane * 33 + c];
    qr[c] = v.x; qi[c] = v.y;
  }

  float* dst = FT + (size_t)b * BATCH_F;
  const float inv = 1.0f / 1024.0f;        // |amp|^2 = 2^-10
  for (int d = 0; d < DA; d++) {
    float rr = 0.0f, ri = 0.0f;
#pragma unroll
    for (int c = 0; c < DA; c++) {
      float2 v = psi[wave][d * 33 + c];    // broadcast read
      rr = fmaf(qr[c],  v.x, rr);
      rr = fmaf(qi[c],  v.y, rr);
      ri = fmaf(qi[c],  v.x, ri);
      ri = fmaf(-qr[c], v.y, ri);
    }
    int k  = lane * DA + d;                // Re feature index
    int k2 = k + 1024;                     // Im feature index
    dst[((size_t)(k  >> 2) * S_ + m) * 4 + (k  & 3)] = rr * inv;
    dst[((size_t)(k2 >> 2) * S_ + m) * 4 + (k2 & 3)] = ri * inv;
  }
}

// ---------------------------------------------------------------------
// K3: fidelity GEMM + fused epilogue.
// Block = 8 waves handles one (batch, 16-row i-tile). Wave w processes
// 64-column chunks cc = w, w+8, ... while 4*cc <= it.
// ---------------------------------------------------------------------
__global__ void __launch_bounds__(256, 1)
attn_kernel(const float* __restrict__ ws,
            float* __restrict__ out) {
  const int b    = blockIdx.y;
  const int it   = blockIdx.x;            // 0..63
  const int wave = threadIdx.x >> 5;      // 0..7
  const int lane = threadIdx.x & 31;
  const int half = lane >> 4;             // fragment half (K / M split)
  const int ln   = lane & 15;             // N index within tile

  const float* Fq = ws + OFF_FQ + (size_t)b * BATCH_F;
  const float* Fk = ws + OFF_FK + (size_t)b * BATCH_F;
  const float* V  = ws + OFF_V  + (size_t)b * S_ * H_;

  __shared__ float ldsZ[8][16];
  __shared__ float ldsN[8][16][H_];

  float zacc[8];
  float nacc[8][H_];
#pragma unroll
  for (int r = 0; r < 8; r++) {
    zacc[r] = 0.0f;
#pragma unroll
    for (int hh = 0; hh < H_; hh++) nacc[r][hh] = 0.0f;
  }

  const int i0 = it * 16;
  const float* abase = Fq + (size_t)(i0 + ln) * 4 + half * 2;
  const size_t kstride = (size_t)S_ * 4;   // floats per K-step

  for (int cc = wave; cc * CH <= it; cc += 8) {
    const int j0 = cc * CH * 16;          // first column of this chunk
    const float* bbase = Fk + (size_t)(j0 + ln) * 4 + half * 2;

    // warm L0 for this wave's NEXT chunk (gfx1250 global_prefetch_b8)
    if ((cc + 8) * CH <= it)
      __builtin_prefetch(Fk + (size_t)((cc + 8) * CH * 16 + ln) * 4, 0, 3);

    v8f cacc[CH];
#pragma unroll
    for (int t = 0; t < CH; t++) cacc[t] = (v8f){};

    // ---- manual 2-stage ping-pong K loop (no buffer rotation) ----
    v2f aA = *(const v2f*)abase;          // stage A holds kb
    v2f bA[CH];
#pragma unroll
    for (int t = 0; t < CH; t++) bA[t] = *(const v2f*)(bbase + t * 64);

#pragma unroll 1
    for (int kb = 0; kb < KB; kb += 2) {
      // issue stage-B loads (kb+1), then compute stage A (kb)
      const float* a1p = abase + (size_t)(kb + 1) * kstride;
      const float* b1p = bbase + (size_t)(kb + 1) * kstride;
      v2f aB = *(const v2f*)a1p;
      v2f bB[CH];
#pragma unroll
      for (int t = 0; t < CH; t++) bB[t] = *(const v2f*)(b1p + t * 64);
#pragma unroll
      for (int t = 0; t < CH; t++) cacc[t] = WMMA_F32(aA, bA[t], cacc[t]);

      // issue stage-A loads (kb+2, clamped), then compute stage B (kb+1)
      const int kb2 = (kb + 2 < KB) ? (kb + 2) : (KB - 1);
      const float* a2p = abase + (size_t)kb2 * kstride;
      const float* b2p = bbase + (size_t)kb2 * kstride;
      aA = *(const v2f*)a2p;
#pragma unroll
      for (int t = 0; t < CH; t++) bA[t] = *(const v2f*)(b2p + t * 64);
#pragma unroll
      for (int t = 0; t < CH; t++) cacc[t] = WMMA_F32(aB, bB[t], cacc[t]);
    }

    // ---- per-chunk epilogue: mask, |a| row sums, a @ V ----
#pragma unroll
    for (int t = 0; t < CH; t++) {
      const int jt = cc * CH + t;
      if (jt > it) break;                 // wave-uniform
      const int jb = j0 + t * 16;
      float vt[H_];
#pragma unroll
      for (int hh = 0; hh < H_; hh++) vt[hh] = V[(jb + ln) * H_ + hh];
#pragma unroll
      for (int r = 0; r < 8; r++) {
        const int gi = i0 + r + 8 * half;
        const int gj = jb + ln;
        const float a = (gj <= gi) ? (2.0f * cacc[t][r] - 1.0f) : 0.0f;
        zacc[r] += fabsf(a);
#pragma unroll
        for (int hh = 0; hh < H_; hh++)
          nacc[r][hh] = fmaf(a, vt[hh], nacc[r][hh]);
      }
    }
  }

  // reduce over n within each 16-lane half (masks stay inside half)
#pragma unroll
  for (int r = 0; r < 8; r++) {
#pragma unroll
    for (int mask = 1; mask < 16; mask <<= 1)
      zacc[r] += __shfl_xor(zacc[r], mask, 32);
#pragma unroll
    for (int hh = 0; hh < H_; hh++)
#pragma unroll
      for (int mask = 1; mask < 16; mask <<= 1)
        nacc[r][hh] += __shfl_xor(nacc[r][hh], mask, 32);
  }

  if (ln == 0) {  // lanes 0 and 16 publish rows 0-7 / 8-15
#pragma unroll
    for (int r = 0; r < 8; r++) {
      int m = r + 8 * half;
      ldsZ[wave][m] = zacc[r];
#pragma unroll
      for (int hh = 0; hh < H_; hh++) ldsN[wave][m][hh] = nacc[r][hh];
    }
  }
  __syncthreads();

  if (threadIdx.x < 16) {
    int m = threadIdx.x;
    float Z = 0.0f;
    float num[H_];
#pragma unroll
    for (int hh = 0; hh < H_; hh++) num[hh] = 0.0f;
#pragma unroll
    for (int w = 0; w < 8; w++) {      // fixed order: deterministic
      Z += ldsZ[w][m];
#pragma unroll
      for (int hh = 0; hh < H_; hh++) num[hh] += ldsN[w][m][hh];
    }
    float invz = 1.0f / fmaxf(Z, 1e-12f);
    int gi = i0 + m;
#pragma unroll
    for (int hh = 0; hh < H_; hh++)
      out[((size_t)b * S_ + gi) * H_ + hh] = num[hh] * invz;
  }
}

// ---------------------------------------------------------------------
extern "C" void kernel_launch(void* const* d_in, const int* in_sizes, int n_in,
                              void* d_out, int out_size, void* d_ws, size_t ws_size,
                              hipStream_t stream) {
  const float* x  = (const float*)d_in[0];
  const float* Wq = (const float*)d_in[1];
  const float* Wk = (const float*)d_in[2];
  const float* Wv = (const float*)d_in[3];
  float* out = (float*)d_out;
  float* ws  = (float*)d_ws;

  // K1: 4096 positions, 4 waves/block
  qkv_kernel<<<dim3((B_ * S_) / 4), 128, 0, stream>>>(x, Wq, Wk, Wv, ws);

  // K2: features for Q and K (one wave per position)
  rho_kernel<<<dim3((B_ * S_) / WPB_B), 32 * WPB_B, 0, stream>>>(
      ws + OFF_Q, ws + OFF_FQ);
  rho_kernel<<<dim3((B_ * S_) / WPB_B), 32 * WPB_B, 0, stream>>>(
      ws + OFF_K, ws + OFF_FK);

  // K3: fused fidelity GEMM + mask + L1 normalize + attn@V
  attn_kernel<<<dim3(S_ / 16, B_), 256, 0, stream>>>(ws, out);
}